// SelfGCN_Block_39676907888770
// MI455X (gfx1250) — compile-verified
//
#include <hip/hip_runtime.h>
#include <hip/hip_bf16.h>
#include <math.h>

// SelfGCN block forward for gfx1250 (MI455X), f32 end-to-end using
// V_WMMA_F32_16X16X4_F32 for every matmul stage.
// Shapes: N=64, C=64, T=128, V=25, OUT=64, REL=8, MID=32.
//
// All tile loads are branchless: unconditional loads at clamped offsets +
// v_cndmask selects; masks are elided at compile time wherever a tile is
// provably dense (tile 0 of the 25-padded dims, full K steps, etc.).

typedef float v2f __attribute__((ext_vector_type(2)));
typedef float v8f __attribute__((ext_vector_type(8)));

#define LANE() (threadIdx.x & 31)

__device__ __forceinline__ v8f wmma4(v2f a, v2f b, v8f c) {
  return __builtin_amdgcn_wmma_f32_16x16x4_f32(
      false, a, false, b, (short)0, c, false, false);
}

__device__ __forceinline__ v8f vzero8() {
  v8f z = {0.f, 0.f, 0.f, 0.f, 0.f, 0.f, 0.f, 0.f};
  return z;
}

// branchless masked scalar load: unconditional load at clamped offset + select
__device__ __forceinline__ float ldsel(const float* __restrict__ p, int off, bool v) {
  float t = p[v ? off : 0];
  return v ? t : 0.f;
}

// ---- tile loaders (wave32 layouts per CDNA5 ISA 7.12.2) -------------------
// A 16x4 (MxK): element (m,k) = base[(m0+m)*lda + k0+k]; pair is contiguous.
template <bool MM, bool MK>
__device__ __forceinline__ v2f load_a(const float* __restrict__ base, int lda,
                                      int m0, int k0, int mMax, int kMax) {
  int l = LANE();
  int m = m0 + (l & 15);
  int k = k0 + ((l >> 4) << 1);
  int o = m * lda + k;
  v2f a;
  if (!MM && !MK) {
    a.x = base[o];
    a.y = base[o + 1];
  } else {
    bool mv = !MM || (m < mMax);
    bool v0 = mv && (!MK || (k < kMax));
    bool v1 = mv && (!MK || (k + 1 < kMax));
    a.x = ldsel(base, o, v0);
    a.y = ldsel(base, o + 1, v1);
  }
  return a;
}

// A 16x4 from transposed source: element (m,k) = base[(k0+k)*ld + m0+m]
template <bool MM, bool MK>
__device__ __forceinline__ v2f load_at(const float* __restrict__ base, int ld,
                                       int m0, int k0, int mMax, int kMax) {
  int l = LANE();
  int m = m0 + (l & 15);
  int k = k0 + ((l >> 4) << 1);
  int o = k * ld + m;
  v2f a;
  if (!MM && !MK) {
    a.x = base[o];
    a.y = base[o + ld];
  } else {
    bool mv = !MM || (m < mMax);
    bool v0 = mv && (!MK || (k < kMax));
    bool v1 = mv && (!MK || (k + 1 < kMax));
    a.x = ldsel(base, o, v0);
    a.y = ldsel(base, o + ld, v1);
  }
  return a;
}

// B 4x16 (KxN): element (k,n) = base[(k0+k)*ldb + n0+n]
template <bool MK, bool MN>
__device__ __forceinline__ v2f load_b(const float* __restrict__ base, int ldb,
                                      int k0, int n0, int kMax, int nMax) {
  int l = LANE();
  int n = n0 + (l & 15);
  int k = k0 + ((l >> 4) << 1);
  int o = k * ldb + n;
  v2f b;
  if (!MK && !MN) {
    b.x = base[o];
    b.y = base[o + ldb];
  } else {
    bool nv = !MN || (n < nMax);
    bool v0 = nv && (!MK || (k < kMax));
    bool v1 = nv && (!MK || (k + 1 < kMax));
    b.x = ldsel(base, o, v0);
    b.y = ldsel(base, o + ldb, v1);
  }
  return b;
}

// B 4x16 from transposed source with scale: B(k,n) = s * base[(n0+n)*ld + k0+k]
template <bool MK, bool MN>
__device__ __forceinline__ v2f load_bt(const float* __restrict__ base, int ld,
                                       int k0, int n0, int kMax, int nMax,
                                       float s) {
  int l = LANE();
  int n = n0 + (l & 15);
  int k = k0 + ((l >> 4) << 1);
  int o = n * ld + k;
  v2f b;
  if (!MK && !MN) {
    b.x = base[o] * s;
    b.y = base[o + 1] * s;
  } else {
    bool nv = !MN || (n < nMax);
    bool v0 = nv && (!MK || (k < kMax));
    bool v1 = nv && (!MK || (k + 1 < kMax));
    b.x = ldsel(base, o, v0) * s;
    b.y = ldsel(base, o + 1, v1) * s;
  }
  return b;
}

// D 16x16 store: element (m,n) -> base[(m0+m)*ldd + n0+n]
template <bool MM, bool MN>
__device__ __forceinline__ void store_d(float* __restrict__ base, int ldd,
                                        int m0, int n0, int mMax, int nMax,
                                        v8f d) {
  int l = LANE();
  int n = n0 + (l & 15);
  int mo = (l >> 4) << 3;
  if (!MN || (n < nMax)) {
#pragma unroll
    for (int r = 0; r < 8; ++r) {
      int m = m0 + mo + r;
      if (!MM || (m < mMax)) base[(size_t)m * ldd + n] = d[r];
    }
  }
}

template <bool MM, bool MN>
__device__ __forceinline__ void store_d_tanh(float* __restrict__ base, int ldd,
                                             int m0, int n0, int mMax, int nMax,
                                             v8f d, float inv) {
  int l = LANE();
  int n = n0 + (l & 15);
  int mo = (l >> 4) << 3;
  if (!MN || (n < nMax)) {
#pragma unroll
    for (int r = 0; r < 8; ++r) {
      int m = m0 + mo + r;
      if (!MM || (m < mMax)) base[(size_t)m * ldd + n] = tanhf(d[r] * inv);
    }
  }
}

// ---------------------------------------------------------------------------
// K0: pack [w1;w2;w11;w22;w3] -> Wp[96,64], biases -> bp[96]
__global__ void k_pack(const float* __restrict__ w1, const float* __restrict__ w2,
                       const float* __restrict__ w11, const float* __restrict__ w22,
                       const float* __restrict__ w3, const float* __restrict__ b1,
                       const float* __restrict__ b2, const float* __restrict__ b11,
                       const float* __restrict__ b22, const float* __restrict__ b3,
                       float* __restrict__ Wp, float* __restrict__ bp) {
  int i = blockIdx.x * blockDim.x + threadIdx.x;
  if (i < 96 * 64) {
    int r = i >> 6, c = i & 63;
    float v;
    if (r < 8)       v = w1[r * 64 + c];
    else if (r < 16) v = w2[(r - 8) * 64 + c];
    else if (r < 24) v = w11[(r - 16) * 64 + c];
    else if (r < 32) v = w22[(r - 24) * 64 + c];
    else             v = w3[(r - 32) * 64 + c];
    Wp[i] = v;
  }
  if (i < 96) {
    bp[i] = (i < 8) ? b1[i] : (i < 16) ? b2[i - 8] : (i < 24) ? b11[i - 16]
            : (i < 32) ? b22[i - 24] : b3[i - 32];
  }
}

// K1: xs[n,0:32,T*V] (x1,x2,x11,x22) and x3[n,0:64,T*V] via GEMM [96,64]x[64,3200]
// wave per (n, otile in 6, ptile in 200); 16 dense WMMA (K=4) each, fully unmasked.
__global__ void k_conv96(const float* __restrict__ x, const float* __restrict__ Wp,
                         const float* __restrict__ bp, float* __restrict__ xs,
                         float* __restrict__ x3) {
  int gw = blockIdx.x * (blockDim.x >> 5) + (threadIdx.x >> 5);
  int pt = gw % 200; int t2 = gw / 200;
  int ot = t2 % 6;   int n  = t2 / 6;
  const float* B = x + (size_t)n * 64 * 3200;   // [64,3200] row-major
  v8f acc = vzero8();
#pragma unroll 4
  for (int k0 = 0; k0 < 64; k0 += 4) {
    v2f a = load_a<false, false>(Wp, 64, ot * 16, k0, 96, 64);
    v2f b = load_b<false, false>(B, 3200, k0, pt * 16, 64, 3200);
    acc = wmma4(a, b, acc);
  }
  int l = LANE();
  int col = pt * 16 + (l & 15);
  int mo = (l >> 4) << 3;
#pragma unroll
  for (int r = 0; r < 8; ++r) {
    int row = ot * 16 + mo + r;
    float v = acc[r] + bp[row];
    if (row < 32)
      xs[(size_t)n * 32 * 3200 + (size_t)row * 3200 + col] = v;
    else
      x3[(size_t)n * 64 * 3200 + (size_t)(row - 32) * 3200 + col] = v;
  }
}

// K2: mean/max over T for x1 (ch 0-7) and x2 (ch 8-15) -> [N,16,V]
__global__ void k_stats(const float* __restrict__ xs, float* __restrict__ smean,
                        float* __restrict__ smax) {
  int idx = blockIdx.x * blockDim.x + threadIdx.x;
  if (idx >= 64 * 16 * 25) return;
  int v = idx % 25; int t2 = idx / 25;
  int c = t2 % 16;  int n = t2 / 16;
  const float* p = xs + (size_t)n * 32 * 3200 + (size_t)c * 3200 + v;
  float s = 0.f, mx = -INFINITY;
  for (int t = 0; t < 128; ++t) { float val = p[t * 25]; s += val; mx = fmaxf(mx, val); }
  smean[idx] = s * (1.f / 128.f);
  smax[idx]  = mx;
}

// K3: x1_res[n,o,u,v] = sum_r w4[o,r]*tanh(w5[r,0]*mean_res + w5[r,1]*max_res + b5[r])*alpha
//                       + b4[o] + A[u,v]
__global__ void k_x1res(const float* __restrict__ smean, const float* __restrict__ smax,
                        const float* __restrict__ w5, const float* __restrict__ b5,
                        const float* __restrict__ w4, const float* __restrict__ b4,
                        const float* __restrict__ Amat, const int* __restrict__ alpha_p,
                        float* __restrict__ x1res) {
  int idx = blockIdx.x * blockDim.x + threadIdx.x;
  if (idx >= 64 * 625) return;
  int v = idx % 25; int u = (idx / 25) % 25; int n = idx / 625;
  float alpha = (float)alpha_p[0];
  float xr8[8];
#pragma unroll
  for (int r = 0; r < 8; ++r) {
    float mr = smean[(n * 16 + r) * 25 + u] - smean[(n * 16 + 8 + r) * 25 + v];
    float xr = smax [(n * 16 + r) * 25 + u] - smax [(n * 16 + 8 + r) * 25 + v];
    xr8[r] = tanhf(w5[r * 2] * mr + w5[r * 2 + 1] * xr + b5[r]) * alpha;
  }
  float av = Amat[u * 25 + v];
  float* out = x1res + (size_t)n * 64 * 625 + u * 25 + v;
  for (int o = 0; o < 64; ++o) {
    float s = b4[o] + av;
#pragma unroll
    for (int r = 0; r < 8; ++r) s += w4[o * 8 + r] * xr8[r];
    out[(size_t)o * 625] = s;
  }
}

// K4: att8[n,c,u,v] = tanh( (1/T) sum_t x1[t,u]*x2[t,v] ); wave per (n,c in 8), K=128
__global__ void k_att8(const float* __restrict__ xs, float* __restrict__ att8) {
  int gw = blockIdx.x * (blockDim.x >> 5) + (threadIdx.x >> 5);  // N*8
  int c = gw % 8; int n = gw / 8;
  const float* x1 = xs + (size_t)n * 32 * 3200 + (size_t)c * 3200;        // [T,V]
  const float* x2 = xs + (size_t)n * 32 * 3200 + (size_t)(8 + c) * 3200;  // [T,V]
  v8f a00 = vzero8(), a01 = vzero8(), a10 = vzero8(), a11 = vzero8();
#pragma unroll 4
  for (int k0 = 0; k0 < 128; k0 += 4) {
    v2f A0 = load_at<false, false>(x1, 25, 0,  k0, 25, 128);  // rows 0-15 dense
    v2f A1 = load_at<true,  false>(x1, 25, 16, k0, 25, 128);  // rows 16-24
    v2f B0 = load_b <false, false>(x2, 25, k0, 0,  128, 25);  // cols 0-15 dense
    v2f B1 = load_b <false, true >(x2, 25, k0, 16, 128, 25);  // cols 16-24
    a00 = wmma4(A0, B0, a00);
    a01 = wmma4(A0, B1, a01);
    a10 = wmma4(A1, B0, a10);
    a11 = wmma4(A1, B1, a11);
  }
  float* out = att8 + (size_t)(n * 8 + c) * 625;
  const float inv = 1.f / 128.f;
  store_d_tanh<false, false>(out, 25, 0,  0,  25, 25, a00, inv);
  store_d_tanh<false, true >(out, 25, 0,  16, 25, 25, a01, inv);
  store_d_tanh<true,  false>(out, 25, 16, 0,  25, 25, a10, inv);
  store_d_tanh<true,  true >(out, 25, 16, 16, 25, 25, a11, inv);
}

// K5: att64[n,o,u,v] = sum_r watt[o,r]*att8[n,r,u,v] + batt[o]
__global__ void k_att64(const float* __restrict__ att8, const float* __restrict__ watt,
                        const float* __restrict__ batt, float* __restrict__ att64) {
  int idx = blockIdx.x * blockDim.x + threadIdx.x;
  if (idx >= 64 * 625) return;
  int p = idx % 625; int n = idx / 625;
  float r8[8];
#pragma unroll
  for (int r = 0; r < 8; ++r) r8[r] = att8[(size_t)(n * 8 + r) * 625 + p];
  float* out = att64 + (size_t)n * 64 * 625 + p;
  for (int o = 0; o < 64; ++o) {
    float s = batt[o];
#pragma unroll
    for (int r = 0; r < 8; ++r) s += watt[o * 8 + r] * r8[r];
    out[(size_t)o * 625] = s;
  }
}

// K6: global_res[n,c] = x3[n,c] (128x25) @ att64[n,c] (25x25); wave per (n,c,mtile in 8)
// 6 dense K-steps (k 0..23) + 1 masked tail (k0=24).
__global__ void k_gres(const float* __restrict__ x3, const float* __restrict__ att64,
                       float* __restrict__ gres) {
  int gw = blockIdx.x * (blockDim.x >> 5) + (threadIdx.x >> 5);  // N*64*8
  int mt = gw % 8; int t2 = gw / 8;
  int c = t2 % 64; int n = t2 / 64;
  const float* A = x3    + (size_t)(n * 64 + c) * 3200;  // [128,25]
  const float* B = att64 + (size_t)(n * 64 + c) * 625;   // [25,25]
  v8f acc0 = vzero8(), acc1 = vzero8();
#pragma unroll
  for (int k0 = 0; k0 < 24; k0 += 4) {
    v2f a  = load_a<false, false>(A, 25, mt * 16, k0, 128, 25);
    v2f b0 = load_b<false, false>(B, 25, k0, 0,  25, 25);
    v2f b1 = load_b<false, true >(B, 25, k0, 16, 25, 25);
    acc0 = wmma4(a, b0, acc0);
    acc1 = wmma4(a, b1, acc1);
  }
  {  // tail k0 = 24 (only k==24 valid)
    v2f a  = load_a<false, true>(A, 25, mt * 16, 24, 128, 25);
    v2f b0 = load_b<true,  false>(B, 25, 24, 0,  25, 25);
    v2f b1 = load_b<true,  true >(B, 25, 24, 16, 25, 25);
    acc0 = wmma4(a, b0, acc0);
    acc1 = wmma4(a, b1, acc1);
  }
  float* out = gres + (size_t)(n * 64 + c) * 3200;
  store_d<false, false>(out, 25, mt * 16, 0,  128, 25, acc0);
  store_d<false, true >(out, 25, mt * 16, 16, 128, 25, acc1);
}

// K7: channel attention: g=mean(gres) -> wc1 -> BN -> GELU -> wc2 -> sigmoid
__global__ void k_catt(const float* __restrict__ gres, const float* __restrict__ wc1,
                       const float* __restrict__ bc1, const float* __restrict__ bng,
                       const float* __restrict__ bnb, const float* __restrict__ bnm,
                       const float* __restrict__ bnv, const float* __restrict__ wc2,
                       const float* __restrict__ bc2, float* __restrict__ catt) {
  __shared__ float part[256];
  __shared__ float g[64];
  __shared__ float h[32];
  int n = blockIdx.x;
  int tid = threadIdx.x;
  int c = tid >> 2, q = tid & 3;
  const float* p = gres + (size_t)(n * 64 + c) * 3200 + q * 800;
  float s = 0.f;
  for (int i = 0; i < 800; ++i) s += p[i];
  part[tid] = s;
  __syncthreads();
  if (q == 0) g[c] = (part[tid] + part[tid + 1] + part[tid + 2] + part[tid + 3]) * (1.f / 3200.f);
  __syncthreads();
  if (tid < 32) {
    float v = bc1[tid];
    for (int c2 = 0; c2 < 64; ++c2) v += wc1[tid * 64 + c2] * g[c2];
    v = (v - bnm[tid]) * (bng[tid] * rsqrtf(bnv[tid] + 1e-5f)) + bnb[tid];
    h[tid] = 0.5f * v * (1.f + erff(v * 0.70710678118654752440f));  // exact GELU
  }
  __syncthreads();
  if (tid < 64) {
    float v = bc2[tid];
    for (int m = 0; m < 32; ++m) v += wc2[tid * 32 + m] * h[m];
    catt[n * 64 + tid] = 1.f / (1.f + expf(-v));
  }
}

// K9: x1r2[n,c] = x3[n,c] (128x25) @ (catt[n,c]*x1res[n,c])^T (25x25)
__global__ void k_x1r2(const float* __restrict__ x3, const float* __restrict__ x1res,
                       const float* __restrict__ catt, float* __restrict__ x1r2) {
  int gw = blockIdx.x * (blockDim.x >> 5) + (threadIdx.x >> 5);  // N*64*8
  int mt = gw % 8; int t2 = gw / 8;
  int c = t2 % 64; int n = t2 / 64;
  float s = catt[n * 64 + c];
  const float* A  = x3    + (size_t)(n * 64 + c) * 3200;  // [128,25]
  const float* Bm = x1res + (size_t)(n * 64 + c) * 625;   // [25,25] (transposed use)
  v8f acc0 = vzero8(), acc1 = vzero8();
#pragma unroll
  for (int k0 = 0; k0 < 24; k0 += 4) {
    v2f a  = load_a <false, false>(A, 25, mt * 16, k0, 128, 25);
    v2f b0 = load_bt<false, false>(Bm, 25, k0, 0,  25, 25, s);
    v2f b1 = load_bt<false, true >(Bm, 25, k0, 16, 25, 25, s);
    acc0 = wmma4(a, b0, acc0);
    acc1 = wmma4(a, b1, acc1);
  }
  {  // tail k0 = 24
    v2f a  = load_a <false, true>(A, 25, mt * 16, 24, 128, 25);
    v2f b0 = load_bt<true,  false>(Bm, 25, 24, 0,  25, 25, s);
    v2f b1 = load_bt<true,  true >(Bm, 25, 24, 16, 25, 25, s);
    acc0 = wmma4(a, b0, acc0);
    acc1 = wmma4(a, b1, acc1);
  }
  float* out = x1r2 + (size_t)(n * 64 + c) * 3200;
  store_d<false, false>(out, 25, mt * 16, 0,  128, 25, acc0);
  store_d<false, true >(out, 25, mt * 16, 16, 128, 25, acc1);
}

// K10: s_att[n,v] = sigmoid( sum_c ws[c] * mean_t x1r2[n,c,t,v] + bs )
__global__ void k_satt(const float* __restrict__ x1r2, const float* __restrict__ wsv,
                       const float* __restrict__ bsv, float* __restrict__ satt) {
  __shared__ float red[256];
  int b = blockIdx.x;  // N*25
  int v = b % 25, n = b / 25;
  int tid = threadIdx.x;
  int c = tid >> 2, tp = tid & 3;
  const float* p = x1r2 + ((size_t)(n * 64 + c) * 128 + tp * 32) * 25 + v;
  float s = 0.f;
  for (int t = 0; t < 32; ++t) s += p[t * 25];
  red[tid] = s * wsv[c];
  __syncthreads();
  for (int st = 128; st > 0; st >>= 1) {
    if (tid < st) red[tid] += red[tid + st];
    __syncthreads();
  }
  if (tid == 0) {
    float z = red[0] * (1.f / 128.f) + bsv[0];
    satt[n * 25 + v] = 1.f / (1.f + expf(-z));
  }
}

// K11: xt[n,t,v,u] = tanh( (1/rel) sum_c x11[c,t,v]*x22[c,t,u] ); wave per (n,t), K=8
__global__ void k_xt(const float* __restrict__ xs, float* __restrict__ xtb) {
  int gw = blockIdx.x * (blockDim.x >> 5) + (threadIdx.x >> 5);  // N*128
  int t = gw % 128; int n = gw / 128;
  const float* x11 = xs + (size_t)n * 32 * 3200 + (size_t)16 * 3200 + t * 25;  // ld 3200
  const float* x22 = xs + (size_t)n * 32 * 3200 + (size_t)24 * 3200 + t * 25;
  v8f a00 = vzero8(), a01 = vzero8(), a10 = vzero8(), a11 = vzero8();
#pragma unroll
  for (int k0 = 0; k0 < 8; k0 += 4) {
    v2f A0 = load_at<false, false>(x11, 3200, 0,  k0, 25, 8);  // A[v,c]
    v2f A1 = load_at<true,  false>(x11, 3200, 16, k0, 25, 8);
    v2f B0 = load_b <false, false>(x22, 3200, k0, 0,  8, 25);  // B[c,u]
    v2f B1 = load_b <false, true >(x22, 3200, k0, 16, 8, 25);
    a00 = wmma4(A0, B0, a00);
    a01 = wmma4(A0, B1, a01);
    a10 = wmma4(A1, B0, a10);
    a11 = wmma4(A1, B1, a11);
  }
  float* out = xtb + (size_t)(n * 128 + t) * 625;
  const float inv = 1.f / 8.f;
  store_d_tanh<false, false>(out, 25, 0,  0,  25, 25, a00, inv);
  store_d_tanh<false, true >(out, 25, 0,  16, 25, 25, a01, inv);
  store_d_tanh<true,  false>(out, 25, 16, 0,  25, 25, a10, inv);
  store_d_tanh<true,  true >(out, 25, 16, 16, 25, 25, a11, inv);
}

// K12: t_res[n,:,t,:] = x3[n,:,t,:] (64x25) @ xt[n,t] (25x25), fused final:
// out = t_res*beita + x1r2 + gres*s_att ; wave per (n,t,ctile in 4)
__global__ void k_final(const float* __restrict__ x3, const float* __restrict__ xtb,
                        const float* __restrict__ x1r2, const float* __restrict__ gres,
                        const float* __restrict__ satt, const float* __restrict__ beita_p,
                        float* __restrict__ out) {
  int gw = blockIdx.x * (blockDim.x >> 5) + (threadIdx.x >> 5);  // N*128*4
  int ct = gw % 4; int t2 = gw / 4;
  int t = t2 % 128; int n = t2 / 128;
  const float* A = x3 + (size_t)n * 64 * 3200 + t * 25;   // A[c,u], ld 3200
  const float* B = xtb + (size_t)(n * 128 + t) * 625;     // B[u,v], ld 25
  v8f acc0 = vzero8(), acc1 = vzero8();
#pragma unroll
  for (int k0 = 0; k0 < 24; k0 += 4) {
    v2f a  = load_a<false, false>(A, 3200, ct * 16, k0, 64, 25);
    v2f b0 = load_b<false, false>(B, 25, k0, 0,  25, 25);
    v2f b1 = load_b<false, true >(B, 25, k0, 16, 25, 25);
    acc0 = wmma4(a, b0, acc0);
    acc1 = wmma4(a, b1, acc1);
  }
  {  // tail k0 = 24
    v2f a  = load_a<false, true>(A, 3200, ct * 16, 24, 64, 25);
    v2f b0 = load_b<true,  false>(B, 25, 24, 0,  25, 25);
    v2f b1 = load_b<true,  true >(B, 25, 24, 16, 25, 25);
    acc0 = wmma4(a, b0, acc0);
    acc1 = wmma4(a, b1, acc1);
  }
  float beita = beita_p[0];
  int l = LANE();
  int mo = (l >> 4) << 3;
  {  // n-tile 0: v in 0..15, always valid
    int v = l & 15;
    float sa = satt[n * 25 + v];
#pragma unroll
    for (int r = 0; r < 8; ++r) {
      int c = ct * 16 + mo + r;
      size_t o = ((size_t)(n * 64 + c) * 128 + t) * 25 + v;
      out[o] = acc0[r] * beita + x1r2[o] + gres[o] * sa;
    }
  }
  {  // n-tile 1: v in 16..31, masked at 25
    int v = 16 + (l & 15);
    if (v < 25) {
      float sa = satt[n * 25 + v];
#pragma unroll
      for (int r = 0; r < 8; ++r) {
        int c = ct * 16 + mo + r;
        size_t o = ((size_t)(n * 64 + c) * 128 + t) * 25 + v;
        out[o] = acc1[r] * beita + x1r2[o] + gres[o] * sa;
      }
    }
  }
}

// ---------------------------------------------------------------------------
extern "C" void kernel_launch(void* const* d_in, const int* in_sizes, int n_in,
                              void* d_out, int out_size, void* d_ws, size_t ws_size,
                              hipStream_t stream) {
  (void)in_sizes; (void)n_in; (void)out_size; (void)ws_size;
  const float* x    = (const float*)d_in[0];
  const float* Amat = (const float*)d_in[1];
  const float* w1   = (const float*)d_in[2];  const float* b1  = (const float*)d_in[3];
  const float* w2   = (const float*)d_in[4];  const float* b2  = (const float*)d_in[5];
  const float* w11  = (const float*)d_in[6];  const float* b11 = (const float*)d_in[7];
  const float* w22  = (const float*)d_in[8];  const float* b22 = (const float*)d_in[9];
  const float* w3   = (const float*)d_in[10]; const float* b3  = (const float*)d_in[11];
  const float* w4   = (const float*)d_in[12]; const float* b4  = (const float*)d_in[13];
  const float* watt = (const float*)d_in[14]; const float* batt= (const float*)d_in[15];
  const float* w5   = (const float*)d_in[16]; const float* b5  = (const float*)d_in[17];
  const float* wc1  = (const float*)d_in[18]; const float* bc1 = (const float*)d_in[19];
  const float* bng  = (const float*)d_in[20]; const float* bnb = (const float*)d_in[21];
  const float* bnm  = (const float*)d_in[22]; const float* bnv = (const float*)d_in[23];
  const float* wc2  = (const float*)d_in[24]; const float* bc2 = (const float*)d_in[25];
  const float* wsv  = (const float*)d_in[26]; const float* bsv = (const float*)d_in[27];
  const float* beita= (const float*)d_in[28];
  const int*   alpha= (const int*)  d_in[29];
  float* outp = (float*)d_out;

  // workspace carve-out (floats); total ~56.5M floats (~226 MB)
  float* w = (float*)d_ws;
  size_t off = 0;
  auto alloc = [&](size_t nfl) { float* p = w + off; off += nfl; return p; };
  float* xs    = alloc((size_t)64 * 32 * 3200);  // x1,x2,x11,x22 stacked
  float* x3    = alloc((size_t)64 * 64 * 3200);
  float* smean = alloc((size_t)64 * 16 * 25);
  float* smax  = alloc((size_t)64 * 16 * 25);
  float* x1res = alloc((size_t)64 * 64 * 625);
  float* att8  = alloc((size_t)64 * 8 * 625);
  float* att64 = alloc((size_t)64 * 64 * 625);
  float* gres  = alloc((size_t)64 * 64 * 3200);
  float* catt  = alloc((size_t)64 * 64);
  float* x1r2  = alloc((size_t)64 * 64 * 3200);
  float* sattb = alloc((size_t)64 * 25);
  float* xtb   = alloc((size_t)64 * 128 * 625);
  float* Wp    = alloc((size_t)96 * 64);
  float* bp    = alloc((size_t)96);

  const int BT = 256;  // 8 wave32s per block
  k_pack  <<<(96 * 64 + BT - 1) / BT, BT, 0, stream>>>(w1, w2, w11, w22, w3,
                                                       b1, b2, b11, b22, b3, Wp, bp);
  k_conv96<<<9600, BT, 0, stream>>>(x, Wp, bp, xs, x3);               // 76800 waves
  k_stats <<<(64 * 16 * 25 + BT - 1) / BT, BT, 0, stream>>>(xs, smean, smax);
  k_x1res <<<(64 * 625 + BT - 1) / BT, BT, 0, stream>>>(smean, smax, w5, b5,
                                                        w4, b4, Amat, alpha, x1res);
  k_att8  <<<64, BT, 0, stream>>>(xs, att8);                          // 512 waves
  k_att64 <<<(64 * 625 + BT - 1) / BT, BT, 0, stream>>>(att8, watt, batt, att64);
  k_gres  <<<4096, BT, 0, stream>>>(x3, att64, gres);                 // 32768 waves
  k_catt  <<<64, BT, 0, stream>>>(gres, wc1, bc1, bng, bnb, bnm, bnv, wc2, bc2, catt);
  k_x1r2  <<<4096, BT, 0, stream>>>(x3, x1res, catt, x1r2);           // 32768 waves
  k_satt  <<<64 * 25, BT, 0, stream>>>(x1r2, wsv, bsv, sattb);
  k_xt    <<<1024, BT, 0, stream>>>(xs, xtb);                         // 8192 waves
  k_final <<<4096, BT, 0, stream>>>(x3, xtb, x1r2, gres, sattb, beita, outp);
}